// BornBeam_13297218748682
// MI455X (gfx1250) — compile-verified
//
#include <hip/hip_runtime.h>
#include <math.h>

// BornBeam on MI455X (gfx1250): batched 256-pt FFTs via radix-16 Cooley-Tukey,
// each radix-16 stage = 16x16 complex matmul on V_WMMA_F32_16X16X4_F32
// (4 chained K=4 WMMAs per real 16x16x16 matmul, 4 real matmuls per complex
//  matmul -> 16 WMMA per stage, 32 per FFT line, one line per wave32).
// DFT/twiddle factors come from tiny per-block LDS tables (W16 has only 16
// distinct values, W256 only 256) instead of per-fragment sincos: v_sin/v_cos
// are quarter-rate.
// FFT input lines are staged into LDS with GLOBAL_LOAD_ASYNC_TO_LDS_B64
// (ASYNCcnt) -- no VGPR round-trip.
//
// Problem sizes (fixed by the reference): D=A=32, H=W=256.
// Scratch layout in d_ws (needs 32 MB):
//   [0,    16MB) : SP = fft2(scatter potential), complex64, [D,H,W]
//   [16MB, 32MB) : F  = per-angle spectral fields, complex64, [A,H,W]

typedef float v2f __attribute__((ext_vector_type(2)));
typedef float v8f __attribute__((ext_vector_type(8)));

#define D_  32
#define A_  32
#define H_  256
#define W_  256
#define HW_ (H_ * W_)
#define PI_F 3.14159265358979323846f

__device__ __forceinline__ v8f wmma_f32(v2f a, v2f b, v8f c) {
  // (neg_a, A, neg_b, B, c_mod, C, reuse_a, reuse_b); F32 WMMA has no A/B
  // negate (ISA: NEG = {CNeg,0,0}) so sign flips are baked into fragments.
  return __builtin_amdgcn_wmma_f32_16x16x4_f32(
      false, a, false, b, (short)0, c, false, false);
}

// Async global->LDS copy of one float2 (B64), per-lane addresses. GV mode:
// 64-bit global address in a VGPR pair, 32-bit LDS byte offset in VDST.
// Tracked by ASYNCcnt; consumer must s_wait_asynccnt 0 first.
__device__ __forceinline__ void async_ld_f2(float2* lds_dst, const float2* g) {
  const unsigned lds_off = (unsigned)(uintptr_t)lds_dst;  // LDS aperture: [31:0]
  asm volatile("global_load_async_to_lds_b64 %0, %1, off"
               :
               : "v"(lds_off), "v"(g)
               : "memory");
}
__device__ __forceinline__ void wait_async0() {
  asm volatile("s_wait_asynccnt 0x0" ::: "memory");
}

// ---------------------------------------------------------------------------
// Scatter potential: SP_real = coef * n * (2*N0 + n), imag = 0.
// ---------------------------------------------------------------------------
__global__ __launch_bounds__(256) void prep_sp(const float* __restrict__ n,
                                               float2* __restrict__ sp,
                                               float coef) {
  const int i = blockIdx.x * 256 + threadIdx.x;
  const float x = n[i];
  sp[i] = make_float2(coef * x * (2.0f * 1.33f + x), 0.0f);
}

// ---------------------------------------------------------------------------
// One 256-point complex FFT per wave (wave32), radix-16 x radix-16 via WMMA.
//   x[n], n = 16*n1 + n2 ; k = k1 + 16*k2
//   stage1: Y[k1,n2] = sum_n1 W16[k1,n1] * x[16 n1 + n2]   (A=W16, B=data)
//   twiddle: T = Y .* W256[k1*n2]
//   stage2: Z[k1,k2] = sum_n2 T[k1,n2] * W16[n2,k2]        (A=T, B=W16)
//   X[k1 + 16 k2] = Z[k1,k2]
// col==0: lines are image rows (stride 1); col==1: columns (stride W).
// sign = -1 forward, +1 inverse; scale applied on output (1/256 per inv pass).
// ---------------------------------------------------------------------------
__global__ __launch_bounds__(128) void fft256_pass(float2* __restrict__ buf,
                                                   int col, float sign,
                                                   float scale) {
  __shared__ float2 xt[4][256];    // per-wave input tile
  __shared__ float2 yt[4][256];    // per-wave inter-stage tile (k1-major)
  __shared__ float2 w16tab[16];    // exp(sign*2pi*i*p/16),  p = (m*k) mod 16
  __shared__ float2 w256tab[256];  // exp(sign*2pi*i*p/256), p = (m*n) mod 256

  const int tid  = threadIdx.x;
  const int wv   = tid >> 5;       // wave id within block (wave32)
  const int lane = tid & 31;
  const int lo   = lane & 15;      // N (or M) index 0..15
  const int hi   = lane >> 4;      // half-wave select

  const int line = blockIdx.x * 4 + wv;
  const int img  = line >> 8;
  const int r    = line & 255;
  const long long base   = (long long)img * HW_ + (col ? r : r * W_);
  const int       stride = col ? W_ : 1;

  // Kick off async global->LDS staging of this wave's 256 complex points.
  for (int j = lane; j < 256; j += 32)
    async_ld_f2(&xt[wv][j], buf + base + (long long)j * stride);

  // Build twiddle tables while the async copies are in flight.
  const float s16  = sign * (2.0f * PI_F / 16.0f);
  const float s256 = sign * (2.0f * PI_F / 256.0f);
  if (tid < 16) {
    float sv, cv;
    __sincosf((float)tid * s16, &sv, &cv);
    w16tab[tid] = make_float2(cv, sv);
  }
  for (int j = tid; j < 256; j += 128) {
    float sv, cv;
    __sincosf((float)j * s256, &sv, &cv);
    w256tab[j] = make_float2(cv, sv);
  }
  wait_async0();
  __syncthreads();

  // ---- stage 1: Y = W16 x Xmat ----
  v8f Yr = {0, 0, 0, 0, 0, 0, 0, 0};
  v8f Yi = {0, 0, 0, 0, 0, 0, 0, 0};
#pragma unroll
  for (int t = 0; t < 4; ++t) {
    const int k0 = 4 * t + 2 * hi;   // A/B fragment K indices (ISA 7.12.2)
    const int k1 = k0 + 1;
    const float2 w0 = w16tab[(lo * k0) & 15];  // W16[M=lo, K=k0]
    const float2 w1 = w16tab[(lo * k1) & 15];
    const v2f ar  = {w0.x, w1.x};
    const v2f ai  = {w0.y, w1.y};
    const v2f nai = {-w0.y, -w1.y};
    const v2f br = {xt[wv][16 * k0 + lo].x, xt[wv][16 * k1 + lo].x};
    const v2f bi = {xt[wv][16 * k0 + lo].y, xt[wv][16 * k1 + lo].y};
    Yr = wmma_f32(ar, br, Yr);   // Re += Wr*Xr
    Yr = wmma_f32(nai, bi, Yr);  // Re -= Wi*Xi
    Yi = wmma_f32(ar, bi, Yi);   // Im += Wr*Xi
    Yi = wmma_f32(ai, br, Yi);   // Im += Wi*Xr
  }

  // ---- twiddle W256^(k1*n2), stash as T[k1][n2] in LDS ----
#pragma unroll
  for (int v = 0; v < 8; ++v) {
    const int M = v + 8 * hi;  // k1   (C/D layout: M = vgpr + 8*(lane/16))
    const int N = lo;          // n2
    const float2 tw = w256tab[(M * N) & 255];
    const float zr = Yr[v] * tw.x - Yi[v] * tw.y;
    const float zi = Yr[v] * tw.y + Yi[v] * tw.x;
    yt[wv][M * 16 + N] = make_float2(zr, zi);
  }
  __syncthreads();

  // ---- stage 2: Z = T x W16 ----
  v8f Zr = {0, 0, 0, 0, 0, 0, 0, 0};
  v8f Zi = {0, 0, 0, 0, 0, 0, 0, 0};
#pragma unroll
  for (int t = 0; t < 4; ++t) {
    const int k0 = 4 * t + 2 * hi;
    const int k1 = k0 + 1;
    const v2f ar = {yt[wv][lo * 16 + k0].x, yt[wv][lo * 16 + k1].x};  // T re
    const v2f ai = {yt[wv][lo * 16 + k0].y, yt[wv][lo * 16 + k1].y};  // T im
    const v2f nai = {-ai[0], -ai[1]};
    const float2 w0 = w16tab[(k0 * lo) & 15];  // W16[K, N=lo]
    const float2 w1 = w16tab[(k1 * lo) & 15];
    const v2f br = {w0.x, w1.x};
    const v2f bi = {w0.y, w1.y};
    Zr = wmma_f32(ar, br, Zr);
    Zr = wmma_f32(nai, bi, Zr);
    Zi = wmma_f32(ar, bi, Zi);
    Zi = wmma_f32(ai, br, Zi);
  }

  // X[k1 + 16*k2] = Z[M=k1, N=k2]
#pragma unroll
  for (int v = 0; v < 8; ++v) {
    const int M = v + 8 * hi;
    const int N = lo;
    buf[base + (long long)(M + 16 * N) * stride] =
        make_float2(Zr[v] * scale, Zi[v] * scale);
  }
}

// ---------------------------------------------------------------------------
// Per (angle, pixel): F = delta + sum_d 0.5i * SP[d, rolled] * E^depth_d / kz,
// then defocus phase exp(-0.5 i kz eva) and binary pupil.
// E^depth evaluated by geometric rotation: P = E^16, P *= conj(E) per step.
// ---------------------------------------------------------------------------
__global__ __launch_bounds__(256) void accum_fields(
    const float2* __restrict__ sp, const float* __restrict__ c_ba,
    float2* __restrict__ F) {
  const int idx = blockIdx.x * 256 + threadIdx.x;
  const int a   = idx >> 16;
  const int hw  = idx & (HW_ - 1);
  const int h   = hw >> 8;
  const int w   = hw & 255;

  // c_gamma: near0(i)/res, near0(i) = fmod(i/256 + 0.5, 1) - 0.5
  const float alpha = (fmodf((float)w * (1.0f / 256.0f) + 0.5f, 1.0f) - 0.5f) * 10.0f;
  const float beta  = (fmodf((float)h * (1.0f / 256.0f) + 0.5f, 1.0f) - 0.5f) * 10.0f;
  const float cg = sqrtf(fmaxf(1.0f - (alpha * alpha + beta * beta), 1e-8f));
  const float kz = cg * (2.0f * PI_F * 0.1f);

  const float cb0 = c_ba[2 * a];
  const float cb1 = c_ba[2 * a + 1];
  const float kzin = sqrtf(1.0f - (cb0 * cb0 + cb1 * cb1)) * (2.0f * PI_F * 0.1f);
  const float defo = kz - kzin;

  // integer rolls: round(c * res * size), RNE like jnp.round
  const int sw = (int)rintf(cb1 * 25.6f);
  const int sh = (int)rintf(cb0 * 25.6f);
  const int hh = (h - sh) & 255;
  const int ww = (w - sw) & 255;

  float er, ei;
  __sincosf(defo, &ei, &er);  // E = exp(i*defo)
  float pr = er, pi_ = ei;    // P = E^16 via 4 squarings
#pragma unroll
  for (int q = 0; q < 4; ++q) {
    const float nr = pr * pr - pi_ * pi_;
    pi_ = 2.0f * pr * pi_;
    pr = nr;
  }
  const float half_over_kz = 0.5f / kz;

  // fields0 = roll(fft2(ones)): H*W at (sh mod H, sw mod W)
  float accr = ((h == (sh & 255)) && (w == (sw & 255))) ? 65536.0f : 0.0f;
  float acci = 0.0f;

  const float2* colp = sp + hh * 256 + ww;
#pragma unroll 4
  for (int d = 0; d < D_; ++d) {
    if (d + 4 < D_) __builtin_prefetch(colp + (d + 4) * HW_, 0, 1);
    const float2 s = colp[d * HW_];
    const float qr = s.x * pr - s.y * pi_;   // s * P
    const float qi = s.x * pi_ + s.y * pr;
    accr -= half_over_kz * qi;               // += 0.5i * q / kz
    acci += half_over_kz * qr;
    const float nr = pr * er + pi_ * ei;     // P *= conj(E)
    const float ni = pi_ * er - pr * ei;
    pr = nr;
    pi_ = ni;
  }

  // diffract(dz=-0.5): phase = exp(-0.5 i * kz * eva)
  const float eva = __expf(fminf((cg - 0.2f) * 5.0f, 0.0f));
  float phs, phc;
  __sincosf(-0.5f * kz * eva, &phs, &phc);
  float fr = accr * phc - acci * phs;
  float fi = accr * phs + acci * phc;

  // binary pupil: cg > sqrt(1 - NA^2)
  const float na_cut = sqrtf(1.0f - 0.65f * 0.65f);
  if (!(cg > na_cut)) { fr = 0.0f; fi = 0.0f; }
  F[idx] = make_float2(fr, fi);
}

// ---------------------------------------------------------------------------
__global__ __launch_bounds__(256) void magsq(const float2* __restrict__ F,
                                             float* __restrict__ out) {
  const int i = blockIdx.x * 256 + threadIdx.x;
  const float2 z = F[i];
  out[i] = z.x * z.x + z.y * z.y;
}

// ---------------------------------------------------------------------------
extern "C" void kernel_launch(void* const* d_in, const int* in_sizes, int n_in,
                              void* d_out, int out_size, void* d_ws,
                              size_t ws_size, hipStream_t stream) {
  (void)in_sizes; (void)n_in; (void)out_size; (void)ws_size;
  const float* n    = (const float*)d_in[0];  // [D,H,W] f32
  const float* c_ba = (const float*)d_in[1];  // [A,2]  f32
  float* out = (float*)d_out;                 // [A,H,W] f32

  float2* sp = (float2*)d_ws;
  float2* F  = (float2*)((char*)d_ws + (size_t)D_ * HW_ * sizeof(float2));

  const float k0   = 2.0f * PI_F * 0.1f / 1.33f;
  const float coef = k0 * k0;  // (2*pi*res0/N0)^2

  // 1) scatter potential (real -> complex)
  prep_sp<<<(D_ * HW_) / 256, 256, 0, stream>>>(n, sp, coef);
  // 2) forward fft2 of SP (in place): rows then columns, sign=-1
  fft256_pass<<<(D_ * H_) / 4, 128, 0, stream>>>(sp, 0, -1.0f, 1.0f);
  fft256_pass<<<(D_ * H_) / 4, 128, 0, stream>>>(sp, 1, -1.0f, 1.0f);
  // 3) per-angle spectral accumulation over depth
  accum_fields<<<(A_ * HW_) / 256, 256, 0, stream>>>(sp, c_ba, F);
  // 4) ifft2 of F (in place): sign=+1, 1/256 per pass
  fft256_pass<<<(A_ * H_) / 4, 128, 0, stream>>>(F, 0, 1.0f, 1.0f / 256.0f);
  fft256_pass<<<(A_ * H_) / 4, 128, 0, stream>>>(F, 1, 1.0f, 1.0f / 256.0f);
  // 5) intensity
  magsq<<<(A_ * HW_) / 256, 256, 0, stream>>>(F, out);
}